// MLPPredictorEGAT_3350074491440
// MI455X (gfx1250) — compile-verified
//
#include <hip/hip_runtime.h>
#include <hip/hip_bf16.h>

typedef float v2f __attribute__((ext_vector_type(2)));
typedef float v8f __attribute__((ext_vector_type(8)));

#define DFEAT 128
#define NOUT  16
#define KTOT  384
#define WPAD  388            // 384 + 4 pad -> stride%64==4 dwords: conflict-free b64 LDS reads
#define WAVES_PER_BLOCK 8

__global__ __launch_bounds__(256) void egat_score_wmma(
    const float* __restrict__ hn, const float* __restrict__ he,
    const int*   __restrict__ src, const int* __restrict__ dst,
    const float* __restrict__ W,  const float* __restrict__ b,
    float* __restrict__ out, int nEdges)
{
    __shared__ float Wlds[NOUT * WPAD];
    __shared__ float blds[NOUT];

    // Stage W (16x384 row-major over k) into padded LDS; W is L2-resident & shared.
    for (int i = threadIdx.x; i < NOUT * KTOT; i += blockDim.x) {
        int n = i / KTOT;
        int k = i - n * KTOT;
        Wlds[n * WPAD + k] = W[i];
    }
    if (threadIdx.x < NOUT) blds[threadIdx.x] = b[threadIdx.x];
    __syncthreads();

    const int lane = threadIdx.x & 31;
    const int wave = threadIdx.x >> 5;
    const int row  = lane & 15;          // A: edge row M ; B: output col N
    const int kh   = (lane >> 4) << 1;   // k-half select: 0 (lanes 0-15) or 2 (lanes 16-31)
    const int hi   = lane >> 4;

    const int nTiles     = nEdges >> 4;
    const int tileStride = gridDim.x * WAVES_PER_BLOCK;

    for (int tile = blockIdx.x * WAVES_PER_BLOCK + wave; tile < nTiles; tile += tileStride) {
        const int e = (tile << 4) + row;
        const int s = src[e];
        const int d = dst[e];
        const float* pu = hn + (size_t)s * DFEAT + kh;   // gathered rows: L2-cached
        const float* pv = hn + (size_t)d * DFEAT + kh;
        const float* pe = he + (size_t)e * DFEAT + kh;   // streamed once: non-temporal
        const float* wrow = &Wlds[row * WPAD + kh];

        v8f c = {0.f, 0.f, 0.f, 0.f, 0.f, 0.f, 0.f, 0.f};

        // segment 0: h_u  (k = 0..127)
        #pragma unroll 8
        for (int k = 0; k < DFEAT; k += 4) {
            v2f a  = *(const v2f*)(pu + k);
            v2f bb = *(const v2f*)(wrow + k);
            c = __builtin_amdgcn_wmma_f32_16x16x4_f32(false, a, false, bb, (short)0, c, false, false);
        }
        // segment 1: h_v  (k = 128..255)
        #pragma unroll 8
        for (int k = 0; k < DFEAT; k += 4) {
            v2f a  = *(const v2f*)(pv + k);
            v2f bb = *(const v2f*)(wrow + DFEAT + k);
            c = __builtin_amdgcn_wmma_f32_16x16x4_f32(false, a, false, bb, (short)0, c, false, false);
        }
        // segment 2: he   (k = 256..383)
        #pragma unroll 8
        for (int k = 0; k < DFEAT; k += 4) {
            v2f a  = __builtin_nontemporal_load((const v2f*)(pe + k));
            v2f bb = *(const v2f*)(wrow + 2 * DFEAT + k);
            c = __builtin_amdgcn_wmma_f32_16x16x4_f32(false, a, false, bb, (short)0, c, false, false);
        }

        // D layout: VGPR r, lanes 0-15 -> M=r ; lanes 16-31 -> M=r+8 ; N = lane&15
        const float bias = blds[row];
        float* orow = out + ((size_t)(tile << 4) + (hi << 3)) * NOUT + row;
        #pragma unroll
        for (int r = 0; r < 8; ++r) {
            __builtin_nontemporal_store(c[r] + bias, orow + (size_t)r * NOUT);
        }
    }
}

// Scalar tail for edges beyond the last full 16-edge tile (not hit when E%16==0).
__global__ void egat_score_tail(
    const float* __restrict__ hn, const float* __restrict__ he,
    const int*   __restrict__ src, const int* __restrict__ dst,
    const float* __restrict__ W,  const float* __restrict__ b,
    float* __restrict__ out, int eStart, int nEdges)
{
    int idx = blockIdx.x * blockDim.x + threadIdx.x;
    int e = eStart + (idx >> 4);
    int o = idx & 15;
    if (e >= nEdges) return;
    const float* pu = hn + (size_t)src[e] * DFEAT;
    const float* pv = hn + (size_t)dst[e] * DFEAT;
    const float* pe = he + (size_t)e * DFEAT;
    const float* w  = W + (size_t)o * KTOT;
    float acc = b[o];
    for (int k = 0; k < DFEAT; ++k) acc += pu[k] * w[k];
    for (int k = 0; k < DFEAT; ++k) acc += pv[k] * w[DFEAT + k];
    for (int k = 0; k < DFEAT; ++k) acc += pe[k] * w[2 * DFEAT + k];
    out[(size_t)e * NOUT + o] = acc;
}

extern "C" void kernel_launch(void* const* d_in, const int* in_sizes, int n_in,
                              void* d_out, int out_size, void* d_ws, size_t ws_size,
                              hipStream_t stream) {
    const float* hn  = (const float*)d_in[0];
    const float* he  = (const float*)d_in[1];
    const int*   src = (const int*)d_in[2];
    const int*   dst = (const int*)d_in[3];
    const float* W   = (const float*)d_in[4];
    const float* b   = (const float*)d_in[5];
    float* out = (float*)d_out;

    const int E = in_sizes[2];          // number of edges
    const int nTiles = E >> 4;

    if (nTiles > 0) {
        int blocks = (nTiles + WAVES_PER_BLOCK - 1) / WAVES_PER_BLOCK;
        if (blocks > 1024) blocks = 1024;   // grid-stride; ~8192 waves in flight
        egat_score_wmma<<<blocks, 256, 0, stream>>>(hn, he, src, dst, W, b, out, E);
    }
    const int rem = E & 15;
    if (rem) {
        int work = rem * NOUT;
        int tb = (work + 255) / 256;
        egat_score_tail<<<tb, 256, 0, stream>>>(hn, he, src, dst, W, b, out, nTiles << 4, E);
    }
}